// DuckDetector_71064528879803
// MI455X (gfx1250) — compile-verified
//
#include <hip/hip_runtime.h>
#include <hip/hip_bf16.h>
#include <stdint.h>

#define BNUM 16
#define NANCH 250000
#define NCLS 10
#define TOPKN 1000
#define PADK 1024
#define TPS 63 /* ceil(1000/16) tiles per side */
#define DETMAX 100
#define SCORE_TH 0.05f
#define NMS_SCORE 0.3f
#define IOU_TH 0.8f
#define NEGV -1000000000.0f
#define BBOX_CLIPV 4.1351665567423560f
#define IMGW 800.0f
#define IMGH 800.0f

typedef float v2f  __attribute__((ext_vector_type(2)));
typedef float v8f  __attribute__((ext_vector_type(8)));
typedef unsigned int u32x4 __attribute__((ext_vector_type(4)));
typedef int i32x4 __attribute__((ext_vector_type(4)));
typedef int i32x8 __attribute__((ext_vector_type(8)));

#if defined(__HIP_DEVICE_COMPILE__) && __has_builtin(__builtin_amdgcn_tensor_load_to_lds)
#define HAVE_TDM 1
#endif

__device__ __forceinline__ unsigned f2key(float f) {
  unsigned u = __float_as_uint(f);
  return (u >> 31) ? ~u : (u | 0x80000000u);
}
__device__ __forceinline__ float key2f(unsigned k) {
  unsigned u = (k & 0x80000000u) ? (k & 0x7FFFFFFFu) : ~k;
  return __uint_as_float(u);
}

// ---------------- init: zero hists, reset radix-select state ----------------
__global__ void k_init(unsigned* hist, unsigned* pfx, unsigned* knd, unsigned* cnt) {
  int g = blockIdx.x * blockDim.x + threadIdx.x;
  if (g < BNUM * 256) hist[g] = 0u;
  if (g < BNUM) { pfx[g] = 0u; knd[g] = (unsigned)TOPKN; cnt[g] = 0u; }
}

// ---------------- score: max-logit -> sigmoid -> sortable key ----------------
__global__ void k_score(const float* __restrict__ logits, unsigned* __restrict__ keys) {
  size_t total = (size_t)BNUM * NANCH;
  size_t stride = (size_t)gridDim.x * blockDim.x;
  for (size_t g = (size_t)blockIdx.x * blockDim.x + threadIdx.x; g < total; g += stride) {
    const float* p = logits + g * NCLS;
    __builtin_prefetch(((const char*)p) + 40960, 0, 0);  // global_prefetch_b8, stream-ahead
    float m = p[0];
#pragma unroll
    for (int c = 1; c < NCLS; ++c) m = fmaxf(m, p[c]);
    float sc = 1.0f / (1.0f + __expf(-m));
    float masked = (sc > SCORE_TH) ? sc : -1.0f;
    keys[g] = f2key(masked);
  }
}

// ---------------- radix-select histogram pass ----------------
__global__ void k_hist(const unsigned* __restrict__ keys, unsigned* __restrict__ hist,
                       const unsigned* __restrict__ pfx, int pass) {
  int b = blockIdx.y;
  __shared__ unsigned h[256];
  for (int i = threadIdx.x; i < 256; i += blockDim.x) h[i] = 0u;
  __syncthreads();
  unsigned prefix = pfx[b];
  int shift = 24 - 8 * pass;
  int cmpsh = shift + 8;
  const unsigned* kb = keys + (size_t)b * NANCH;
  for (int i = blockIdx.x * blockDim.x + threadIdx.x; i < NANCH; i += gridDim.x * blockDim.x) {
    unsigned k = kb[i];
    bool ok = (pass == 0) ? true : ((k >> cmpsh) == (prefix >> cmpsh));
    if (ok) atomicAdd(&h[(k >> shift) & 255u], 1u);
  }
  __syncthreads();
  for (int i = threadIdx.x; i < 256; i += blockDim.x)
    if (h[i]) atomicAdd(&hist[b * 256 + i], h[i]);
}

// ---------------- radix-select pivot pick (device-side, no host readback) ----
__global__ void k_pick(unsigned* hist, unsigned* pfx, unsigned* knd, unsigned* thr, int pass) {
  int b = blockIdx.x, t = threadIdx.x;
  __shared__ unsigned h[256];
  h[t] = hist[b * 256 + t];
  hist[b * 256 + t] = 0u;  // re-zero for next pass
  __syncthreads();
  if (t == 0) {
    unsigned prefix = pfx[b], kneed = knd[b];
    int shift = 24 - 8 * pass;
    unsigned chosen = 0;
    for (int byte = 255; byte >= 0; --byte) {
      unsigned c = h[byte];
      if (c < kneed) { kneed -= c; }
      else { chosen = (unsigned)byte; break; }
    }
    prefix |= chosen << shift;
    pfx[b] = prefix; knd[b] = kneed;
    if (pass == 3) thr[b] = prefix;  // exact 32-bit k-th threshold
  }
}

// ---------------- compact candidates >= threshold ----------------
__global__ void k_compact(const unsigned* __restrict__ keys, const unsigned* __restrict__ thr,
                          unsigned* cnt, unsigned long long* cand) {
  int b = blockIdx.y;
  unsigned th = thr[b];
  const unsigned* kb = keys + (size_t)b * NANCH;
  for (int i = blockIdx.x * blockDim.x + threadIdx.x; i < NANCH; i += gridDim.x * blockDim.x) {
    unsigned k = kb[i];
    if (k >= th) {
      unsigned pos = atomicAdd(&cnt[b], 1u);
      if (pos < PADK) cand[(size_t)b * PADK + pos] = ((unsigned long long)k << 32) | (unsigned)i;
    }
  }
}

// ---------------- sort candidates (TDM -> LDS -> bitonic), decode boxes ------
__global__ void __launch_bounds__(PADK)
k_sort_decode(const unsigned long long* __restrict__ cand, const unsigned* __restrict__ cnt,
              const float* __restrict__ reg, const float* __restrict__ anc,
              const float* __restrict__ logits,
              float* __restrict__ boxes, float* __restrict__ tsc, unsigned* __restrict__ labs) {
  __shared__ unsigned long long scsh[PADK];
  int b = blockIdx.x, t = threadIdx.x;
  const unsigned long long* candb = cand + (size_t)b * PADK;

#ifdef HAVE_TDM
  if (t < 32) {  // wave 0 issues one TDM descriptor: 1024x8B tile, global -> LDS
    unsigned long long ga = (unsigned long long)(uintptr_t)candb;
    unsigned ldsa = (unsigned)(uintptr_t)(&scsh[0]);
    u32x4 g0; g0[0] = 1u;                      // count=1, user descriptor
    g0[1] = ldsa;                              // lds_addr
    g0[2] = (unsigned)(ga & 0xFFFFFFFFu);      // global_addr[31:0]
    g0[3] = (unsigned)((ga >> 32) & 0x1FFFFFFu) | 0x80000000u;  // addr[56:32] | type=2
    i32x8 g1; g1[0] = 0x30000;                 // data_size=3 (8B), no multicast
    g1[1] = 0x04000000;                        // tensor_dim0 lo16=1024 in [31:16]
    g1[2] = 0x00010000;                        // tensor_dim1=1
    g1[3] = 0x04000000;                        // tile_dim0=1024 in [31:16]
    g1[4] = 1;                                 // tile_dim1=1
    g1[5] = 1024;                              // tensor_dim0_stride=1024
    g1[6] = 0; g1[7] = 0;
    i32x4 z4 = {0, 0, 0, 0};
#if defined(__clang_major__) && __clang_major__ >= 23
    i32x8 z8 = {0, 0, 0, 0, 0, 0, 0, 0};
    __builtin_amdgcn_tensor_load_to_lds(g0, g1, z4, z4, z8, 0);
#else
    __builtin_amdgcn_tensor_load_to_lds(g0, g1, z4, z4, 0);
#endif
#if __has_builtin(__builtin_amdgcn_s_wait_tensorcnt)
    __builtin_amdgcn_s_wait_tensorcnt(0);
#endif
  }
  __syncthreads();
#else
  scsh[t] = candb[t];
  __syncthreads();
#endif

  unsigned c = cnt[b]; if (c > PADK) c = PADK;
  if ((unsigned)t >= c) scsh[t] = 0ull;  // pad unfilled slots

  // bitonic sort, descending on (key<<32|idx)
  for (unsigned k = 2; k <= PADK; k <<= 1) {
    for (unsigned j = k >> 1; j > 0; j >>= 1) {
      __syncthreads();
      unsigned ixj = (unsigned)t ^ j;
      if (ixj > (unsigned)t) {
        unsigned long long a = scsh[t], bb = scsh[ixj];
        if ((a < bb) == (((unsigned)t & k) == 0u)) { scsh[t] = bb; scsh[ixj] = a; }
      }
    }
  }
  __syncthreads();

  size_t row = (size_t)b * PADK + t;
  if (t < TOPKN) {
    unsigned long long e = scsh[t];
    unsigned key = (unsigned)(e >> 32);
    unsigned idx = (unsigned)e;
    float s = key2f(key);
    const float4* ap = (const float4*)anc; const float4* rp = (const float4*)reg;
    float4 A = ap[(size_t)b * NANCH + idx];
    float4 R = rp[(size_t)b * NANCH + idx];
    float w = A.z - A.x, h = A.w - A.y;
    float cx = A.x + 0.5f * w, cy = A.y + 0.5f * h;
    float dw = fminf(R.z, BBOX_CLIPV), dh = fminf(R.w, BBOX_CLIPV);
    float pcx = R.x * w + cx, pcy = R.y * h + cy;
    float pw = __expf(dw) * w, ph = __expf(dh) * h;
    float x1 = fminf(fmaxf(pcx - 0.5f * pw, 0.f), IMGW);
    float y1 = fminf(fmaxf(pcy - 0.5f * ph, 0.f), IMGH);
    float x2 = fminf(fmaxf(pcx + 0.5f * pw, 0.f), IMGW);
    float y2 = fminf(fmaxf(pcy + 0.5f * ph, 0.f), IMGH);
    boxes[row * 4 + 0] = x1; boxes[row * 4 + 1] = y1;
    boxes[row * 4 + 2] = x2; boxes[row * 4 + 3] = y2;
    tsc[row] = s;
    const float* lp = logits + ((size_t)b * NANCH + idx) * NCLS;
    float m = lp[0]; unsigned lab = 0;
#pragma unroll
    for (int cc = 1; cc < NCLS; ++cc) { float v = lp[cc]; if (v > m) { m = v; lab = cc; } }
    labs[row] = lab;
  } else {
    boxes[row * 4 + 0] = 0.f; boxes[row * 4 + 1] = 0.f;
    boxes[row * 4 + 2] = 0.f; boxes[row * 4 + 3] = 0.f;
    tsc[row] = NEGV; labs[row] = 0xFFFFFFFFu;
  }
}

// ---------------- pairwise IoU matrix, area-sum tiles via WMMA f32 16x16x4 ---
__global__ void k_iou(const float* __restrict__ boxes, float* __restrict__ iou) {
  int b = blockIdx.y;
  int wid = threadIdx.x >> 5, lane = threadIdx.x & 31;
  int tile = blockIdx.x * (blockDim.x >> 5) + wid;
  int maxt = TPS * TPS - 1;
  tile = (tile > maxt) ? maxt : tile;  // clamp (duplicate work) to keep barriers uniform
  int ti = tile / TPS, tj = tile % TPS;
  const float* Bx = boxes + (size_t)b * PADK * 4;
  __shared__ float rb[8][16][4];
  __shared__ float cb[8][16][4];
  if (lane < 16) {
    int r = ti * 16 + lane;
#pragma unroll
    for (int q = 0; q < 4; ++q) rb[wid][lane][q] = Bx[r * 4 + q];
  } else {
    int cI = tj * 16 + (lane - 16);
#pragma unroll
    for (int q = 0; q < 4; ++q) cb[wid][lane - 16][q] = Bx[cI * 4 + q];
  }
  __syncthreads();

  // A(16x4) rows = [area_i, 1, 0, 0]; B(4x16) cols = [1, area_j, 0, 0]^T
  // => C_ij = area_i + area_j (the union term), rank-2 matmul on the WMMA unit.
  v2f a, bv;
  if (lane < 16) {
    const float* q = rb[wid][lane];
    const float* p = cb[wid][lane];
    a.x = (q[2] - q[0]) * (q[3] - q[1]); a.y = 1.0f;   // K=0,1
    bv.x = 1.0f; bv.y = (p[2] - p[0]) * (p[3] - p[1]);
  } else {
    a.x = 0.f; a.y = 0.f; bv.x = 0.f; bv.y = 0.f;      // K=2,3 contribute zero
  }
  v8f cacc;
#pragma unroll
  for (int q = 0; q < 8; ++q) cacc[q] = 0.0f;
  cacc = __builtin_amdgcn_wmma_f32_16x16x4_f32(false, a, false, bv, (short)0, cacc,
                                               false, false);

  int colL = lane & 15, hs = (lane >> 4) * 8;
#pragma unroll
  for (int v = 0; v < 8; ++v) {
    int rr = v + hs;  // C layout: VGPR v -> M = v + 8*(lane>=16), N = lane&15
    const float* q = rb[wid][rr];
    const float* p = cb[wid][colL];
    float ix1 = fmaxf(q[0], p[0]), iy1 = fmaxf(q[1], p[1]);
    float ix2 = fminf(q[2], p[2]), iy2 = fminf(q[3], p[3]);
    float inter = fmaxf(ix2 - ix1, 0.f) * fmaxf(iy2 - iy1, 0.f);
    float uni = cacc[v] - inter + 1e-9f;
    int gr = ti * 16 + rr, gc = tj * 16 + colL;
    iou[((size_t)b * PADK + gr) * PADK + gc] = inter / uni;
  }
}

// ---------------- soft-NMS + sort + greedy suppression (1 WG / image) --------
__global__ void __launch_bounds__(PADK)
k_nms(const float* __restrict__ iou, const float* __restrict__ boxes,
      const float* __restrict__ tsc, const unsigned* __restrict__ labs,
      float* __restrict__ out) {
  int b = blockIdx.x, t = threadIdx.x, lane = t & 31, wid = t >> 5;
  __shared__ float s[PADK];
  __shared__ unsigned lab[PADK];
  __shared__ unsigned char proc[PADK];
  __shared__ float selS[PADK];
  __shared__ unsigned short selI[PADK];
  __shared__ unsigned char keptSel[PADK];
  __shared__ float wv[32];
  __shared__ int wi[32];
  __shared__ int bcastI;
  __shared__ unsigned long long sk[PADK];
  __shared__ unsigned short msel[PADK];
  __shared__ unsigned char keptR[PADK];
  __shared__ unsigned char alive[PADK];
  __shared__ unsigned char fin[PADK];
  __shared__ int scan[PADK];
  __shared__ float outS[PADK];

  s[t] = (t < TOPKN) ? tsc[(size_t)b * PADK + t] : NEGV;
  lab[t] = labs[(size_t)b * PADK + t];
  proc[t] = 0;
  __syncthreads();

  const float* ioub = iou + (size_t)b * PADK * PADK;

  // ---- soft-NMS: 1000 sequential steps; precomputed IoU rows off critical path
  for (int i = 0; i < TOPKN; ++i) {
    float v = (t < TOPKN && !proc[t]) ? s[t] : NEGV;
    int id = t;
#pragma unroll
    for (int off = 16; off; off >>= 1) {  // wave32 argmax
      float ov = __shfl_xor(v, off, 32);
      int oi = __shfl_xor(id, off, 32);
      if (ov > v || (ov == v && oi < id)) { v = ov; id = oi; }
    }
    if (lane == 0) { wv[wid] = v; wi[wid] = id; }
    __syncthreads();
    if (wid == 0) {
      float v2 = wv[lane]; int i2 = wi[lane];
#pragma unroll
      for (int off = 16; off; off >>= 1) {
        float ov = __shfl_xor(v2, off, 32);
        int oi = __shfl_xor(i2, off, 32);
        if (ov > v2 || (ov == v2 && oi < i2)) { v2 = ov; i2 = oi; }
      }
      if (lane == 0) {
        bcastI = i2; proc[i2] = 1;
        selI[i] = (unsigned short)i2; selS[i] = v2;
      }
    }
    __syncthreads();
    int bi = bcastI;
    if (t < TOPKN && !proc[t] && lab[t] == lab[bi]) {
      float io = ioub[(size_t)bi * PADK + t];         // coalesced row read
      s[t] *= __expf(-2.0f * io * io);                // exp(-iou^2 / 0.5)
    }
    __syncthreads();
  }

  // ---- sort selections by (kept ? score : NEG) descending
  if (t < TOPKN) {
    unsigned char kp = selS[t] > NMS_SCORE;
    keptSel[t] = kp;
    float scv = kp ? selS[t] : NEGV;
    sk[t] = ((unsigned long long)f2key(scv) << 32) | (unsigned)t;
  } else {
    sk[t] = 0ull;
  }
  for (unsigned k = 2; k <= PADK; k <<= 1) {
    for (unsigned j = k >> 1; j > 0; j >>= 1) {
      __syncthreads();
      unsigned ixj = (unsigned)t ^ j;
      if (ixj > (unsigned)t) {
        unsigned long long aa = sk[t], bb = sk[ixj];
        if ((aa < bb) == (((unsigned)t & k) == 0u)) { sk[t] = bb; sk[ixj] = aa; }
      }
    }
  }
  __syncthreads();
  if (t < TOPKN) {
    unsigned j = (unsigned)sk[t];
    outS[t] = key2f((unsigned)(sk[t] >> 32));
    keptR[t] = keptSel[j];
    msel[t] = selI[j];
    alive[t] = 1; fin[t] = 0;
  } else {
    alive[t] = 0; fin[t] = 0; msel[t] = 0; keptR[t] = 0; outS[t] = 0.f;
  }
  __syncthreads();

  // ---- greedy suppression (IoU via permuted lookups into precomputed matrix)
  int mycol = msel[(t < TOPKN) ? t : 0];
  for (int i = 0; i < TOPKN; ++i) {
    int cond = alive[i] && keptR[i];
    int rowm = msel[i];
    __syncthreads();
    if (cond && t < TOPKN) {
      float io = ioub[(size_t)rowm * PADK + mycol];
      if (io > IOU_TH) alive[t] = 0;
    }
    if (t == 0) { alive[i] = 0; fin[i] = (unsigned char)cond; }
    __syncthreads();
  }

  // ---- rank = inclusive cumsum(final); cap DETMAX
  scan[t] = (t < TOPKN) ? (int)fin[t] : 0;
  __syncthreads();
  for (int off = 1; off < PADK; off <<= 1) {
    int v = (t >= off) ? scan[t - off] : 0;
    __syncthreads();
    scan[t] += v;
    __syncthreads();
  }

  if (t < TOPKN) {
    int f = fin[t] && (scan[t] <= DETMAX);
    float x1 = 0.f, y1 = 0.f, x2 = 0.f, y2 = 0.f, sc2 = 0.f;
    if (f) {
      const float4* bp = (const float4*)boxes;
      float4 bb4 = bp[(size_t)b * PADK + msel[t]];
      x1 = bb4.x; y1 = bb4.y; x2 = bb4.z; y2 = bb4.w;
      sc2 = outS[t];
    }
    float* o = out + ((size_t)b * TOPKN + t) * 5;
    o[0] = x1; o[1] = y1; o[2] = x2; o[3] = y2; o[4] = sc2;
  }
}

extern "C" void kernel_launch(void* const* d_in, const int* in_sizes, int n_in,
                              void* d_out, int out_size, void* d_ws, size_t ws_size,
                              hipStream_t stream) {
  (void)in_sizes; (void)n_in; (void)out_size; (void)ws_size;
  const float* logits = (const float*)d_in[0];
  const float* reg    = (const float*)d_in[1];
  const float* anc    = (const float*)d_in[2];
  float* out = (float*)d_out;

  char* ws = (char*)d_ws;
  size_t off = 0;
  unsigned* keys = (unsigned*)(ws + off); off += (size_t)BNUM * NANCH * 4;   // 16.0 MB
  unsigned* hist = (unsigned*)(ws + off); off += (size_t)BNUM * 256 * 4;
  unsigned* pfx  = (unsigned*)(ws + off); off += (size_t)BNUM * 4;
  unsigned* knd  = (unsigned*)(ws + off); off += (size_t)BNUM * 4;
  unsigned* thr  = (unsigned*)(ws + off); off += (size_t)BNUM * 4;
  unsigned* cnt  = (unsigned*)(ws + off); off += (size_t)BNUM * 4;
  off = (off + 255) & ~(size_t)255;
  unsigned long long* cand = (unsigned long long*)(ws + off); off += (size_t)BNUM * PADK * 8;
  float* boxes = (float*)(ws + off); off += (size_t)BNUM * PADK * 4 * 4;
  float* tsc   = (float*)(ws + off); off += (size_t)BNUM * PADK * 4;
  unsigned* labs = (unsigned*)(ws + off); off += (size_t)BNUM * PADK * 4;
  off = (off + 255) & ~(size_t)255;
  float* ioum = (float*)(ws + off); off += (size_t)BNUM * PADK * PADK * 4;   // 64 MB

  k_init<<<BNUM, 256, 0, stream>>>(hist, pfx, knd, cnt);
  k_score<<<4096, 256, 0, stream>>>(logits, keys);
  for (int p = 0; p < 4; ++p) {
    k_hist<<<dim3(128, BNUM), 256, 0, stream>>>(keys, hist, pfx, p);
    k_pick<<<BNUM, 256, 0, stream>>>(hist, pfx, knd, thr, p);
  }
  k_compact<<<dim3(128, BNUM), 256, 0, stream>>>(keys, thr, cnt, cand);
  k_sort_decode<<<BNUM, PADK, 0, stream>>>(cand, cnt, reg, anc, logits, boxes, tsc, labs);
  k_iou<<<dim3((TPS * TPS + 7) / 8, BNUM), 256, 0, stream>>>(boxes, ioum);
  k_nms<<<BNUM, PADK, 0, stream>>>(ioum, boxes, tsc, labs, out);
}